// Controller_66425964199939
// MI455X (gfx1250) — compile-verified
//
#include <hip/hip_runtime.h>

#define D 2048
#define V 4096

typedef __attribute__((ext_vector_type(2))) float v2f;
typedef __attribute__((ext_vector_type(4))) float v4f;
typedef __attribute__((ext_vector_type(8))) float v8f;
typedef __attribute__((ext_vector_type(4))) int v4i;

// ---------------------------------------------------------------------------
// Async global->LDS staging (CDNA5 GLOBAL_LOAD_ASYNC_TO_LDS_B128, ASYNCcnt),
// guarded so the file still compiles if the builtin is absent.
// Probe-derived signature: (AS1 v4i* src, AS3 v4i* dst, imm int off, imm cpol).
// ---------------------------------------------------------------------------
#if defined(__has_builtin)
#if __has_builtin(__builtin_amdgcn_global_load_async_to_lds_b128)
#define HAVE_ASYNC_LDS 1
#endif
#endif
#ifndef HAVE_ASYNC_LDS
#define HAVE_ASYNC_LDS 0
#endif

#if HAVE_ASYNC_LDS
typedef __attribute__((address_space(1))) v4i as1_v4i;
typedef __attribute__((address_space(3))) v4i as3_v4i;
#endif

__device__ __forceinline__ void stage16(const float* __restrict__ g,
                                        float* __restrict__ s, int fidx) {
  // copy floats [4*fidx, 4*fidx+4) from global to LDS
#if HAVE_ASYNC_LDS
  __builtin_amdgcn_global_load_async_to_lds_b128((as1_v4i*)(g + 4 * fidx),
                                                 (as3_v4i*)(s + 4 * fidx), 0,
                                                 0);
#else
  *(v4f*)(s + 4 * fidx) = *(const v4f*)(g + 4 * fidx);
#endif
}

__device__ __forceinline__ void async_join() {
#if HAVE_ASYNC_LDS
#if defined(__has_builtin) && __has_builtin(__builtin_amdgcn_s_wait_asynccnt)
  __builtin_amdgcn_s_wait_asynccnt(0);
#else
  asm volatile("s_wait_asynccnt 0x0" ::: "memory");
#endif
#endif
  __syncthreads();
}

// ---------------------------------------------------------------------------
// f32 WMMA helper: D = A(16x4) * B(4x16) + C(16x16), full f32 precision.
// A/B layout (ISA 7.12.2, 32-bit): lanes 0-15 = M/N rows, K = vgpr + 2*(lane>=16),
// so each lane's fragment is a contiguous float2 from its row.
// ---------------------------------------------------------------------------
__device__ __forceinline__ v8f wmma_k4(v2f a, v2f b, v8f c) {
  return __builtin_amdgcn_wmma_f32_16x16x4_f32(false, a, false, b, (short)0, c,
                                               false, false);
}

// A fragment: only row M=0 is live (the GEMV vector); mask with selects, no
// exec-branching in the hot loop.
__device__ __forceinline__ v2f a_frag(const float* __restrict__ shx, int kk,
                                      int hi2, bool alane) {
  const v2f xa = *(const v2f*)(shx + kk + hi2);
  v2f a;
  a.x = alane ? xa.x : 0.f;
  a.y = alane ? xa.y : 0.f;
  return a;
}

// ---------------------------------------------------------------------------
// Generic GEMV: y[r] = bias[r] + sum_k W[r][k] * x[k],  W row-major [R, K].
// grid.x = R/16, block = 256 (8 waves, each covers K/8). x = xtab + sel[0]*K.
// K must equal D (true for every use here).
// ---------------------------------------------------------------------------
__global__ void gemv16_wmma(const float* __restrict__ xtab,
                            const int* __restrict__ sel,
                            const float* __restrict__ W,
                            const float* __restrict__ bias,
                            float* __restrict__ y, int K) {
  __shared__ __align__(16) float shx[D];
  const float* x = xtab + (size_t)sel[0] * (size_t)K;
  const int tid = threadIdx.x;
  stage16(x, shx, tid);
  stage16(x, shx, tid + 256);

  const int n0 = blockIdx.x * 16;
  const int wave = tid >> 5, lane = tid & 31;
  const int kper = K >> 3, k0 = wave * kper;
  const int hi2 = (lane >> 4) << 1;  // 0 or 2
  const float* wrow = W + (size_t)(n0 + (lane & 15)) * (size_t)K;
  const bool alane = (lane & 15) == 0;

  async_join();

  v8f acc = {0.f, 0.f, 0.f, 0.f, 0.f, 0.f, 0.f, 0.f};
#pragma unroll 4
  for (int kk = k0; kk < k0 + kper; kk += 4) {
    const v2f a = a_frag(shx, kk, hi2, alane);
    const v2f b = *(const v2f*)(wrow + kk + hi2);
    acc = wmma_k4(a, b, acc);
  }
  // Row M=0 of C lives in vgpr0 of lanes 0..15 (N = lane).
  __shared__ float part[8][16];
  if (lane < 16) part[wave][lane] = acc[0];
  __syncthreads();
  if (tid < 16) {
    float s = bias[n0 + tid];
#pragma unroll
    for (int w = 0; w < 8; ++w) s += part[w][tid];
    y[n0 + tid] = s;
  }
}

// ---------------------------------------------------------------------------
// LSTM gate pre-activations: gates[r] = b_ih[r]+b_hh[r] + W_ih[r]·x + W_hh[r]·h
// grid.x = 8192/16 = 512, block = 256 (8 waves split K).
// ---------------------------------------------------------------------------
__global__ void lstm_gates_wmma(const float* __restrict__ xtab,
                                const int* __restrict__ xsel,
                                const float* __restrict__ h_in,
                                const float* __restrict__ w_ih,
                                const float* __restrict__ w_hh,
                                const float* __restrict__ b_ih,
                                const float* __restrict__ b_hh,
                                float* __restrict__ gates) {
  __shared__ __align__(16) float shx[D];
  __shared__ __align__(16) float shh[D];
  const float* x = xtab + (size_t)xsel[0] * (size_t)D;
  const int tid = threadIdx.x;
  stage16(x, shx, tid);
  stage16(x, shx, tid + 256);
  stage16(h_in, shh, tid);
  stage16(h_in, shh, tid + 256);

  const int n0 = blockIdx.x * 16;
  const int wave = tid >> 5, lane = tid & 31;
  const int kper = D >> 3, k0 = wave * kper;
  const int hi2 = (lane >> 4) << 1;
  const size_t row = (size_t)(n0 + (lane & 15));
  const float* wi = w_ih + row * (size_t)D;
  const float* wh = w_hh + row * (size_t)D;
  const bool alane = (lane & 15) == 0;

  async_join();

  v8f acc = {0.f, 0.f, 0.f, 0.f, 0.f, 0.f, 0.f, 0.f};
#pragma unroll 4
  for (int kk = k0; kk < k0 + kper; kk += 4) {
    const v2f a = a_frag(shx, kk, hi2, alane);
    const v2f b = *(const v2f*)(wi + kk + hi2);
    acc = wmma_k4(a, b, acc);
  }
#pragma unroll 4
  for (int kk = k0; kk < k0 + kper; kk += 4) {
    const v2f a = a_frag(shh, kk, hi2, alane);
    const v2f b = *(const v2f*)(wh + kk + hi2);
    acc = wmma_k4(a, b, acc);
  }
  __shared__ float part[8][16];
  if (lane < 16) part[wave][lane] = acc[0];
  __syncthreads();
  if (tid < 16) {
    const int r = n0 + tid;
    float s = b_ih[r] + b_hh[r];
#pragma unroll
    for (int w = 0; w < 8; ++w) s += part[w][tid];
    gates[r] = s;
  }
}

// ---------------------------------------------------------------------------
// LSTM pointwise update (gate order i,f,g,o). Ping-pong buffers avoid races.
// ---------------------------------------------------------------------------
__global__ void lstm_pointwise(const float* __restrict__ gates,
                               const float* __restrict__ c_in,
                               float* __restrict__ h_out,
                               float* __restrict__ c_out,
                               float* __restrict__ c_copy) {
  const int t = blockIdx.x * blockDim.x + threadIdx.x;
  if (t >= D) return;
  const float ig = 1.f / (1.f + expf(-gates[t]));
  const float fg = 1.f / (1.f + expf(-gates[D + t]));
  const float gg = tanhf(gates[2 * D + t]);
  const float og = 1.f / (1.f + expf(-gates[3 * D + t]));
  const float cn = fg * c_in[t] + ig * gg;
  const float hn = og * tanhf(cn);
  c_out[t] = cn;
  h_out[t] = hn;
  if (c_copy) c_copy[t] = cn;
}

// ---------------------------------------------------------------------------
// Attention logits: logits[j] = lin3_b + sum_k tanh(q2[k]+all_cw[j][k])*lin3_w[k]
// ---------------------------------------------------------------------------
__global__ void skip_logits_kernel(const float* __restrict__ q2,
                                   const float* __restrict__ all_cw, int nq,
                                   const float* __restrict__ lin3_w,
                                   const float* __restrict__ lin3_b,
                                   float* __restrict__ logits) {
  __shared__ float red[256];
  const int t = threadIdx.x;
  for (int j = 0; j < nq; ++j) {
    float s = 0.f;
    for (int k = t; k < D; k += 256)
      s += tanhf(q2[k] + all_cw[j * D + k]) * lin3_w[k];
    red[t] = s;
    __syncthreads();
    for (int o = 128; o > 0; o >>= 1) {
      if (t < o) red[t] += red[t + o];
      __syncthreads();
    }
    if (t == 0) logits[j] = red[0] + lin3_b[0];
    __syncthreads();
  }
}

// ---------------------------------------------------------------------------
// JAX threefry2x32 (matches jax._src.prng.threefry2x32).
// ---------------------------------------------------------------------------
__device__ __forceinline__ unsigned rotl32(unsigned x, int r) {
  return (x << r) | (x >> (32 - r));
}
__device__ __forceinline__ void threefry2x32(unsigned k0, unsigned k1,
                                             unsigned x0, unsigned x1,
                                             unsigned& o0, unsigned& o1) {
  const unsigned ks2 = k0 ^ k1 ^ 0x1BD11BDAu;
  x0 += k0; x1 += k1;
#define TF_R4(a, b, c, d)                                    \
  x0 += x1; x1 = rotl32(x1, a); x1 ^= x0;                    \
  x0 += x1; x1 = rotl32(x1, b); x1 ^= x0;                    \
  x0 += x1; x1 = rotl32(x1, c); x1 ^= x0;                    \
  x0 += x1; x1 = rotl32(x1, d); x1 ^= x0;
  TF_R4(13, 15, 26, 6)  x0 += k1;  x1 += ks2 + 1u;
  TF_R4(17, 29, 16, 24) x0 += ks2; x1 += k0 + 2u;
  TF_R4(13, 15, 26, 6)  x0 += k0;  x1 += k1 + 3u;
  TF_R4(17, 29, 16, 24) x0 += k1;  x1 += ks2 + 4u;
  TF_R4(13, 15, 26, 6)  x0 += ks2; x1 += k0 + 5u;
#undef TF_R4
  o0 = x0; o1 = x1;
}

// random bits for flat element i of an n-element draw (JAX counter split +
// odd-size zero padding): counts [0..n-1] (+pad 0), halves are the 2x32 block.
__device__ __forceinline__ unsigned jax_random_bits(unsigned fk0, unsigned fk1,
                                                    int i, int n) {
  const int half = (n + 1) >> 1;
  const bool lo = i < half;
  const int j = lo ? i : (i - half);
  const unsigned c0 = (unsigned)j;
  const int p = half + j;
  const unsigned c1 = (p < n) ? (unsigned)p : 0u;
  unsigned o0, o1;
  threefry2x32(fk0, fk1, c0, c1, o0, o1);
  return lo ? o0 : o1;
}

__device__ __forceinline__ float gumbel_from_bits(unsigned bits) {
  const float tiny = 1.1754943508222875e-38f;  // finfo(f32).tiny
  const unsigned fb = (bits >> 9) | 0x3f800000u;
  const float u01 = __builtin_bit_cast(float, fb) - 1.0f;  // [0,1)
  float u = fmaxf(u01 * (1.0f - tiny) + tiny, tiny);       // [tiny,1)
  return -logf(-logf(u));
}

// ---------------------------------------------------------------------------
// sample(): categorical via Gumbel-argmax with JAX PRNG; also entropy + logp.
// out layout: [0..8]=arc (as float), [9..17]=entropy, [18..26]=log_probs.
// ---------------------------------------------------------------------------
__global__ void sample_kernel(const float* __restrict__ logits, int n, int step,
                              int slot, float* __restrict__ out,
                              int* __restrict__ sel) {
  __shared__ float s_m[256], s_v[256], s_a[256], s_b[256];
  __shared__ int s_i[256];
  const int t = threadIdx.x;
  unsigned fk0, fk1;  // fold_in(key(42), step) = threefry(key=(0,42), (0,step))
  threefry2x32(0u, 42u, 0u, (unsigned)step, fk0, fk1);

  float lmax = -INFINITY, bv = -INFINITY;
  int bi = 0x7fffffff;
  for (int i = t; i < n; i += 256) {
    const float l = logits[i];
    const float v = l + gumbel_from_bits(jax_random_bits(fk0, fk1, i, n));
    lmax = fmaxf(lmax, l);
    if (v > bv || (v == bv && i < bi)) { bv = v; bi = i; }
  }
  s_m[t] = lmax; s_v[t] = bv; s_i[t] = bi;
  __syncthreads();
  for (int o = 128; o > 0; o >>= 1) {
    if (t < o) {
      s_m[t] = fmaxf(s_m[t], s_m[t + o]);
      const float v2 = s_v[t + o]; const int i2 = s_i[t + o];
      if (v2 > s_v[t] || (v2 == s_v[t] && i2 < s_i[t])) { s_v[t] = v2; s_i[t] = i2; }
    }
    __syncthreads();
  }
  const float m = s_m[0];
  const int idx = s_i[0];
  float se = 0.f, sl = 0.f;
  for (int i = t; i < n; i += 256) {
    const float e = expf(logits[i] - m);
    se += e; sl += e * logits[i];
  }
  s_a[t] = se; s_b[t] = sl;
  __syncthreads();
  for (int o = 128; o > 0; o >>= 1) {
    if (t < o) { s_a[t] += s_a[t + o]; s_b[t] += s_b[t + o]; }
    __syncthreads();
  }
  if (t == 0) {
    const float S = s_a[0], L = s_b[0];
    const float logZ = m + logf(S);
    out[slot] = (float)idx;
    out[9 + slot] = logZ - L / S;          // entropy
    out[18 + slot] = logits[idx] - logZ;   // selected log-prob
    sel[0] = idx;
  }
}

__global__ void init_kernel(float* __restrict__ h0, float* __restrict__ c0,
                            int* __restrict__ zeroi) {
  const int t = blockIdx.x * blockDim.x + threadIdx.x;
  if (t < D) { h0[t] = 0.f; c0[t] = 0.f; }
  if (t == 0) zeroi[0] = 0;
}

// ---------------------------------------------------------------------------
extern "C" void kernel_launch(void* const* d_in, const int* in_sizes, int n_in,
                              void* d_out, int out_size, void* d_ws,
                              size_t ws_size, hipStream_t stream) {
  (void)in_sizes; (void)n_in; (void)out_size; (void)ws_size;
  const float* emb0   = (const float*)d_in[0];
  const float* w_ih   = (const float*)d_in[1];
  const float* w_hh   = (const float*)d_in[2];
  const float* b_ih   = (const float*)d_in[3];
  const float* b_hh   = (const float*)d_in[4];
  const float* lin1_w = (const float*)d_in[5];
  const float* lin1_b = (const float*)d_in[6];
  const float* lin2_w = (const float*)d_in[7];
  const float* lin2_b = (const float*)d_in[8];
  const float* lin3_w = (const float*)d_in[9];
  const float* lin3_b = (const float*)d_in[10];
  const float* dec_w  = (const float*)d_in[11];
  const float* dec_b  = (const float*)d_in[12];
  const float* emb_t  = (const float*)d_in[13];
  float* out = (float*)d_out;

  // workspace layout (floats)
  float* ws     = (float*)d_ws;
  float* hbuf   = ws;             // 2*D (ping-pong h)
  float* cbuf   = ws + 2 * D;     // 2*D (ping-pong c)
  float* all_c  = ws + 4 * D;     // 4*D
  float* all_cw = ws + 8 * D;     // 4*D
  float* q2     = ws + 12 * D;    // D
  float* gates  = ws + 13 * D;    // 4*D
  float* logits = ws + 17 * D;    // V
  int* ibuf  = (int*)(ws + 17 * D + V);
  int* zeroi = ibuf;              // constant 0 index
  int* selb  = ibuf + 1;          // 9 sampled indices

  init_kernel<<<8, 256, 0, stream>>>(hbuf, cbuf, zeroi);

  int pp = 0;  // current (h,c) buffer
  const float* xtab = emb0;
  const int* xsel = zeroi;
  int step = 0;

  for (int layer = 0; layer < 4; ++layer) {
    // LSTM cell #1
    lstm_gates_wmma<<<512, 256, 0, stream>>>(xtab, xsel, hbuf + pp * D, w_ih,
                                             w_hh, b_ih, b_hh, gates);
    lstm_pointwise<<<8, 256, 0, stream>>>(gates, cbuf + pp * D,
                                          hbuf + (1 - pp) * D,
                                          cbuf + (1 - pp) * D,
                                          all_c + layer * D);
    pp = 1 - pp;
    const float* ccur = cbuf + pp * D;
    // all_cw[layer] = c @ lin1_w.T + lin1_b ; q2 = c @ lin2_w.T + lin2_b
    gemv16_wmma<<<128, 256, 0, stream>>>(ccur, zeroi, lin1_w, lin1_b,
                                         all_cw + layer * D, D);
    gemv16_wmma<<<128, 256, 0, stream>>>(ccur, zeroi, lin2_w, lin2_b, q2, D);
    skip_logits_kernel<<<1, 256, 0, stream>>>(q2, all_cw, layer + 1, lin3_w,
                                              lin3_b, logits);
    sample_kernel<<<1, 256, 0, stream>>>(logits, layer + 1, step, 2 * layer,
                                         out, selb + 2 * layer);
    ++step;

    // LSTM cell #2, input = all_c[skip_idx]
    lstm_gates_wmma<<<512, 256, 0, stream>>>(all_c, selb + 2 * layer,
                                             hbuf + pp * D, w_ih, w_hh, b_ih,
                                             b_hh, gates);
    lstm_pointwise<<<8, 256, 0, stream>>>(gates, cbuf + pp * D,
                                          hbuf + (1 - pp) * D,
                                          cbuf + (1 - pp) * D, nullptr);
    pp = 1 - pp;
    ccur = cbuf + pp * D;
    gemv16_wmma<<<256, 256, 0, stream>>>(ccur, zeroi,
                                         dec_w + (size_t)layer * V * D,
                                         dec_b + layer * V, logits, D);
    sample_kernel<<<1, 256, 0, stream>>>(logits, V, step, 2 * layer + 1, out,
                                         selb + 2 * layer + 1);
    ++step;

    xtab = emb_t + (size_t)layer * V * D;
    xsel = selb + 2 * layer + 1;
  }

  // final decision
  lstm_gates_wmma<<<512, 256, 0, stream>>>(xtab, xsel, hbuf + pp * D, w_ih,
                                           w_hh, b_ih, b_hh, gates);
  lstm_pointwise<<<8, 256, 0, stream>>>(gates, cbuf + pp * D,
                                        hbuf + (1 - pp) * D,
                                        cbuf + (1 - pp) * D, nullptr);
  pp = 1 - pp;
  gemv16_wmma<<<256, 256, 0, stream>>>(cbuf + pp * D, zeroi,
                                       dec_w + (size_t)4 * V * D, dec_b + 4 * V,
                                       logits, D);
  sample_kernel<<<1, 256, 0, stream>>>(logits, V, step, 8, out, selb + 8);
}